// FHNNeuralODEModel_61143154426011
// MI455X (gfx1250) — compile-verified
//
#include <hip/hip_runtime.h>

typedef __attribute__((ext_vector_type(2))) float v2f;
typedef __attribute__((ext_vector_type(8))) float v8f;

// Compiler-only ordering fence: single-wave LDS is hardware in-order (DScnt
// decrements in order), so no s_barrier / s_wait_*cnt drain is needed — we
// only need to stop the compiler from reordering LDS accesses.
#define LDS_COMPILER_FENCE() asm volatile("" ::: "memory")

__device__ __forceinline__ float fast_tanh(float x) {
#if __has_builtin(__builtin_amdgcn_tanhf)
    return __builtin_amdgcn_tanhf(x);          // v_tanh_f32 (gfx1250 HW trans)
#elif __has_builtin(__builtin_amdgcn_tanh_f32)
    return __builtin_amdgcn_tanh_f32(x);
#else
    return tanhf(x);
#endif
}

#if __has_builtin(__builtin_amdgcn_update_dpp)
// x + dpp_perm(x); DPP row ops act within 16-lane rows (exactly our duplicated
// half-waves), so after xor1,xor2,ror4,ror8 every lane holds the 16-col sum.
#define DPP_ADD(x, ctrl)                                                      \
    ((x) + __builtin_bit_cast(float, __builtin_amdgcn_update_dpp(             \
               0, __builtin_bit_cast(int, (x)), (ctrl), 0xf, 0xf, true)))
__device__ __forceinline__ float row_reduce_bcast(float x) {
    x = DPP_ADD(x, 0xB1);   // quad_perm [1,0,3,2]  (xor 1)
    x = DPP_ADD(x, 0x4E);   // quad_perm [2,3,0,1]  (xor 2)
    x = DPP_ADD(x, 0x124);  // row_ror:4
    x = DPP_ADD(x, 0x128);  // row_ror:8
    return x;
}
#else
__device__ __forceinline__ float row_reduce_bcast(float x) {
    for (int m = 1; m <= 8; m <<= 1) x += __shfl_xor(x, m, 32);
    return x;
}
#endif

// Single-wave latency kernel: sequential RK4(3/8) neural-ODE rollout.
// 64x64 layer via V_WMMA_F32_16X16X4_F32 (f32 matrix path, wave32).
__global__ __launch_bounds__(32)
void fhn_ode_wmma_kernel(const float* __restrict__ t,
                         const float* __restrict__ v0,
                         const float* __restrict__ W1,
                         const float* __restrict__ b1,
                         const float* __restrict__ W2,
                         const float* __restrict__ b2,
                         const float* __restrict__ W3,
                         const float* __restrict__ b3,
                         const float* __restrict__ w0,
                         float* __restrict__ out,
                         int T)
{
    const int lane = threadIdx.x;   // 0..31 (wave32)
    const int col  = lane & 15;     // N column within a 16-wide tile
    const int half = lane >> 4;     // 0 -> K pair {0,1}; 1 -> K pair {2,3}

    __shared__ float hs[64];        // staging for layer-1 activations

    // ---- One-time weight preload into registers ----
    // W2 in B-operand layout for wmma_f32_16x16x4_f32:
    //   K-slab s (k=4s..4s+3), N-tile tt (n=16tt..16tt+15):
    //   lane(half,col) holds B[k][n] = W2[n][k], k = 4s+2*half+{0,1}, n = 16tt+col.
    v2f wb[16][4];
#pragma unroll
    for (int s = 0; s < 16; ++s) {
#pragma unroll
        for (int tt = 0; tt < 4; ++tt) {
            const float* p = W2 + (16 * tt + col) * 64 + 4 * s + 2 * half;
            wb[s][tt] = {p[0], p[1]};
        }
    }

    // Layer-1: lane owns rows 2*lane and 2*lane+1 of W1 (64x2) and b1.
    const float w1a0 = W1[(2 * lane) * 2 + 0];
    const float w1a1 = W1[(2 * lane) * 2 + 1];
    const float b1a  = b1[2 * lane];
    const float w1b0 = W1[(2 * lane + 1) * 2 + 0];
    const float w1b1 = W1[(2 * lane + 1) * 2 + 1];
    const float b1b  = b1[2 * lane + 1];

    // Layer-2 bias and layer-3 weights for the output columns this lane owns.
    float b2t[4], w3a[4], w3b[4];
#pragma unroll
    for (int tt = 0; tt < 4; ++tt) {
        b2t[tt] = b2[16 * tt + col];
        w3a[tt] = W3[0 * 64 + 16 * tt + col];
        w3b[tt] = W3[1 * 64 + 16 * tt + col];
    }
    const float b30 = b3[0];
    const float b31 = b3[1];

    // Initial state y = [v0, w0], redundant in every lane.
    float y0 = v0[0];
    float y1 = w0[0];
    if (lane == 0) { out[0] = y0; out[1] = y1; }

    const float third = 1.0f / 3.0f;

    // MLP vector field: (k0,k1) = f(yy0,yy1); result broadcast to all lanes.
    auto f = [&](float yy0, float yy1, float& k0, float& k1) {
        // ---- layer 1 (2 -> 64), two rows per lane ----
        float pa = fmaf(w1a0, yy0, fmaf(w1a1, yy1, b1a));
        float pb = fmaf(w1b0, yy0, fmaf(w1b1, yy1, b1b));
        float ha = fast_tanh(pa);
        float hb = fast_tanh(pb);
        LDS_COMPILER_FENCE();                 // prev readers ordered before store
        hs[2 * lane]     = ha;
        hs[2 * lane + 1] = hb;
        LDS_COMPILER_FENCE();                 // store ordered before loads

        // ---- gather A-operand K-slabs (A replicated across M) ----
        v2f a[16];
#pragma unroll
        for (int s = 0; s < 16; ++s) {
            const float* p = &hs[4 * s + 2 * half];
            a[s] = {p[0], p[1]};
        }

        // ---- layer 2 (64 -> 64): 8 interleaved WMMA chains of depth 8 ----
        const v8f zero = {0.f, 0.f, 0.f, 0.f, 0.f, 0.f, 0.f, 0.f};
        v8f accA[4], accB[4];
#pragma unroll
        for (int tt = 0; tt < 4; ++tt) {
            accA[tt] = __builtin_amdgcn_wmma_f32_16x16x4_f32(
                false, a[0], false, wb[0][tt], (short)0, zero, false, false);
            accB[tt] = __builtin_amdgcn_wmma_f32_16x16x4_f32(
                false, a[8], false, wb[8][tt], (short)0, zero, false, false);
        }
#pragma unroll
        for (int s = 1; s < 8; ++s) {
#pragma unroll
            for (int tt = 0; tt < 4; ++tt) {
                accA[tt] = __builtin_amdgcn_wmma_f32_16x16x4_f32(
                    false, a[s], false, wb[s][tt], (short)0, accA[tt], false, false);
                accB[tt] = __builtin_amdgcn_wmma_f32_16x16x4_f32(
                    false, a[s + 8], false, wb[s + 8][tt], (short)0, accB[tt], false, false);
            }
        }

        // D replicated across M: component 0 holds out[16*tt + col].
        float h2[4];
#pragma unroll
        for (int tt = 0; tt < 4; ++tt)
            h2[tt] = fast_tanh(accA[tt][0] + accB[tt][0] + b2t[tt]);

        // ---- layer 3 (64 -> 2): per-lane partials + DPP row reduction ----
        float p0 = 0.0f, p1 = 0.0f;
#pragma unroll
        for (int tt = 0; tt < 4; ++tt) {
            p0 = fmaf(w3a[tt], h2[tt], p0);
            p1 = fmaf(w3b[tt], h2[tt], p1);
        }
        k0 = row_reduce_bcast(p0) + b30;
        k1 = row_reduce_bcast(p1) + b31;
    };

    // ---- sequential RK4 (3/8 rule) rollout ----
    float tprev = t[0];
    for (int i = 0; i + 1 < T; ++i) {
        float tnext = t[i + 1];               // issued early; overlaps k1 eval
        float k10, k11, k20, k21, k30, k31, k40, k41;
        f(y0, y1, k10, k11);
        const float dt = tnext - tprev;
        tprev = tnext;
        f(fmaf(dt * third, k10, y0), fmaf(dt * third, k11, y1), k20, k21);
        f(fmaf(dt, k20 - k10 * third, y0), fmaf(dt, k21 - k11 * third, y1), k30, k31);
        f(fmaf(dt, k10 - k20 + k30, y0), fmaf(dt, k11 - k21 + k31, y1), k40, k41);
        const float w = dt * 0.125f;
        y0 = fmaf(w, k10 + 3.0f * (k20 + k30) + k40, y0);
        y1 = fmaf(w, k11 + 3.0f * (k21 + k31) + k41, y1);
        if (lane == 0) {
            out[2 * i + 2] = y0;
            out[2 * i + 3] = y1;
        }
    }
}

extern "C" void kernel_launch(void* const* d_in, const int* in_sizes, int n_in,
                              void* d_out, int out_size, void* d_ws, size_t ws_size,
                              hipStream_t stream) {
    const float* t  = (const float*)d_in[0];
    const float* v0 = (const float*)d_in[1];
    const float* W1 = (const float*)d_in[2];
    const float* b1 = (const float*)d_in[3];
    const float* W2 = (const float*)d_in[4];
    const float* b2 = (const float*)d_in[5];
    const float* W3 = (const float*)d_in[6];
    const float* b3 = (const float*)d_in[7];
    const float* w0 = (const float*)d_in[8];
    const int T = in_sizes[0];

    hipLaunchKernelGGL(fhn_ode_wmma_kernel, dim3(1), dim3(32), 0, stream,
                       t, v0, W1, b1, W2, b2, W3, b3, w0, (float*)d_out, T);
}